// PointCloudMeshGridLoss_79070347920145
// MI455X (gfx1250) — compile-verified
//
#include <hip/hip_runtime.h>
#include <math.h>

// ---------------------------------------------------------------------------
// Point-to-mesh loss for MI455X (gfx1250, wave32).
//
// loss_b = sqrt( sum_p min_f dist^2(p, tri_f) ) / height_b ; out = mean_b.
//
// Per (point,tri) pair the Ericson closest-point test needs three dot products
// with the point:  s = ab.p, t = ac.p, q = |p|^2 - 2 a.p.  We compute those
// for 16 points x 16 triangles with three V_WMMA_F32_16X16X4_F32 ops (exact
// fp32).  The VALU epilogue selects one (vnum, wnum, den) triple per region
// and does a single v_rcp_f32 per pair (no IEEE-div expansion).
// ---------------------------------------------------------------------------

typedef __attribute__((ext_vector_type(2))) float v2f;
typedef __attribute__((ext_vector_type(8))) float v8f;

#define B_  2
#define P_  8192
#define V_  4098
#define F_  8192
#define SPLIT 8            // face-dimension split for occupancy
#define TRI_STRIDE 20      // floats per triangle record (80B, float4 aligned)
#define EPSF 1e-12f
#define BIGF 3.402823466e38f

__device__ __forceinline__ float fdot3(float ax,float ay,float az,
                                       float bx,float by,float bz){
    return fmaf(ax,bx,fmaf(ay,by,az*bz));
}

// ---------------- init: minBuf = +FLT_MAX (as uint bits) -------------------
__global__ void init_k(unsigned* __restrict__ minBuf, int n){
    int i = blockIdx.x*blockDim.x + threadIdx.x;
    if (i < n) minBuf[i] = 0x7F7FFFFFu;  // FLT_MAX bit pattern
}

// ---------------- triangle preprocessing -----------------------------------
// record: [ab(3) ac(3) a(3) ab.a ac.a ab.b ac.b ab.c ac.c |a|^2 AA AB CC pad]
__global__ void prep_tri(const float* __restrict__ verts,
                         const int*   __restrict__ faces,
                         float* __restrict__ tri){
    int idx = blockIdx.x*blockDim.x + threadIdx.x;      // b*F + f
    if (idx >= B_*F_) return;
    int b = idx / F_;
    const float* vb = verts + (size_t)b * V_ * 3;
    const int*   fc = faces + (size_t)idx * 3;
    int i0 = fc[0], i1 = fc[1], i2 = fc[2];
    float ax=vb[i0*3+0], ay=vb[i0*3+1], az=vb[i0*3+2];
    float bx=vb[i1*3+0], by=vb[i1*3+1], bz=vb[i1*3+2];
    float cx=vb[i2*3+0], cy=vb[i2*3+1], cz=vb[i2*3+2];
    float abx=bx-ax, aby=by-ay, abz=bz-az;
    float acx=cx-ax, acy=cy-ay, acz=cz-az;
    float* t = tri + (size_t)idx * TRI_STRIDE;
    t[0]=abx; t[1]=aby; t[2]=abz;
    t[3]=acx; t[4]=acy; t[5]=acz;
    t[6]=ax;  t[7]=ay;  t[8]=az;
    t[9]  = fdot3(abx,aby,abz, ax,ay,az);   // ab.a
    t[10] = fdot3(acx,acy,acz, ax,ay,az);   // ac.a
    t[11] = fdot3(abx,aby,abz, bx,by,bz);   // ab.b
    t[12] = fdot3(acx,acy,acz, bx,by,bz);   // ac.b
    t[13] = fdot3(abx,aby,abz, cx,cy,cz);   // ab.c
    t[14] = fdot3(acx,acy,acz, cx,cy,cz);   // ac.c
    t[15] = fdot3(ax,ay,az, ax,ay,az);      // |a|^2
    t[16] = fdot3(abx,aby,abz, abx,aby,abz);// AA
    t[17] = fdot3(abx,aby,abz, acx,acy,acz);// AB
    t[18] = fdot3(acx,acy,acz, acx,acy,acz);// CC
    t[19] = 0.f;
}

// ---------------- per-batch mesh height ------------------------------------
__global__ void height_k(const float* __restrict__ verts, float* __restrict__ heights){
    __shared__ float smx[128], smn[128];
    int b = blockIdx.x, tid = threadIdx.x;
    const float* vb = verts + (size_t)b * V_ * 3;
    float mx = -BIGF, mn = BIGF;
    for (int i = tid; i < V_; i += 128){
        float y = vb[i*3+1];
        mx = fmaxf(mx, y); mn = fminf(mn, y);
    }
    smx[tid]=mx; smn[tid]=mn; __syncthreads();
    for (int s=64; s>0; s>>=1){
        if (tid < s){ smx[tid]=fmaxf(smx[tid],smx[tid+s]); smn[tid]=fminf(smn[tid],smn[tid+s]); }
        __syncthreads();
    }
    if (tid==0) heights[b] = smx[0]-smn[0];
}

// ---------------- main WMMA distance kernel --------------------------------
// grid (P/16, SPLIT, B), block = one wave32.
__global__ void __launch_bounds__(32)
dist_k(const float* __restrict__ points,
       const float* __restrict__ tri,
       unsigned*    __restrict__ minBuf){
    const int lane  = threadIdx.x;       // 0..31
    const int m     = lane & 15;         // row (point) / col (triangle) slot
    const int hi    = lane >> 4;         // half-wave selector
    const int ptile = blockIdx.x;
    const int split = blockIdx.y;
    const int b     = blockIdx.z;

    // A matrix row for point m: (px, py, pz, |p|^2).
    // 16x4 fp32 A layout: lanes 0-15 hold K=0,1 in vgpr0,1; lanes 16-31 hold K=2,3.
    const float* pp = points + ((size_t)b*P_ + (size_t)ptile*16 + m)*3;
    float px = pp[0], py = pp[1], pz = pp[2];
    float p2 = fdot3(px,py,pz,px,py,pz);
    v2f A;
    A.x = hi ? pz : px;
    A.y = hi ? p2 : py;

    float minD[8];
#pragma unroll
    for (int r=0;r<8;++r) minD[r] = BIGF;

    const int tilesPerSplit = F_/16/SPLIT;           // 64
    const int tile0 = split * tilesPerSplit;
    const float* triB = tri + (size_t)b*F_*TRI_STRIDE;

    for (int tt = 0; tt < tilesPerSplit; ++tt){
        const int trIdx = (tile0+tt)*16 + m;
        const float4* tc = (const float4*)(triB + (size_t)trIdx*TRI_STRIDE);
        float4 c0 = tc[0];   // abx aby abz acx
        float4 c1 = tc[1];   // acy acz ax  ay
        float4 c2 = tc[2];   // az  ab.a ac.a ab.b
        float4 c3 = tc[3];   // ac.b ab.c ac.c |a|^2
        float4 c4 = tc[4];   // AA AB CC pad
        if (tt+1 < tilesPerSplit)   // -> global_prefetch_b8
            __builtin_prefetch(triB + (size_t)((tile0+tt+1)*16 + m)*TRI_STRIDE, 0, 0);

        // B columns (4x16): S uses (ab,0), T uses (ac,0), U uses (-2a,1).
        v2f bS, bT, bU;
        bS.x = hi ? c0.z : c0.x;        bS.y = hi ? 0.f : c0.y;
        bT.x = hi ? c1.y : c0.w;        bT.y = hi ? 0.f : c1.x;
        bU.x = hi ? -2.f*c2.x : -2.f*c1.z;
        bU.y = hi ? 1.f       : -2.f*c1.w;

        v8f Z = {0.f,0.f,0.f,0.f,0.f,0.f,0.f,0.f};
        v8f S = __builtin_amdgcn_wmma_f32_16x16x4_f32(false, A, false, bS, (short)0, Z, false, false);
        v8f T = __builtin_amdgcn_wmma_f32_16x16x4_f32(false, A, false, bT, (short)0, Z, false, false);
        v8f Q = __builtin_amdgcn_wmma_f32_16x16x4_f32(false, A, false, bU, (short)0, Z, false, false);

        const float cABa=c2.y, cACa=c2.z, cABb=c2.w, cACb=c3.x;
        const float cABc=c3.y, cACc=c3.z, nA=c3.w;
        const float AAc=c4.x, ABc=c4.y, CCc=c4.z;

#pragma unroll
        for (int r=0;r<8;++r){
            float s = S[r], t = T[r];
            float papsq = Q[r] + nA;                 // |p-a|^2
            float d1 = s - cABa, d2 = t - cACa;
            float d3 = s - cABb, d4 = t - cACb;
            float d5 = s - cABc, d6 = t - cACc;
            float va = d3*d6 - d5*d4;
            float vb = d5*d2 - d1*d6;
            float vc = d1*d4 - d3*d2;
            float e43 = d4 - d3, e56 = d5 - d6;

            // Region classification (reference where-chain priority: later wins).
            bool r6 = (va<=0.f) && (e43>=0.f) && (e56>=0.f);   // edge bc
            bool r5 = (vb<=0.f) && (d2>=0.f) && (d6<=0.f);     // edge ac
            bool r4 = (vc<=0.f) && (d1>=0.f) && (d3<=0.f);     // edge ab
            bool r3 = (d6>=0.f) && (d5<=d6);                   // vertex c
            bool r2 = (d3>=0.f) && (d4<=d3);                   // vertex b
            bool r1 = (d1<=0.f) && (d2<=0.f);                  // vertex a

            // One deferred division: (v,w) = (vnum,wnum)/den for the winning region.
            float vnum = vb, wnum = vc, den = va+vb+vc;        // interior
            if (r6){ vnum = e56; wnum = e43; den = e43+e56; }
            if (r5){ vnum = 0.f; wnum = d2;  den = d2-d6; }
            if (r4){ vnum = d1;  wnum = 0.f; den = d1-d3; }
            float rden = __builtin_amdgcn_rcpf(fmaxf(den, EPSF));  // v_rcp_f32
            float v = vnum*rden, w = wnum*rden;
            if (r3){ v = 0.f; w = 1.f; }
            if (r2){ v = 1.f; w = 0.f; }
            if (r1){ v = 0.f; w = 0.f; }

            // |p - (a + v*ab + w*ac)|^2
            float dd = papsq + v*fmaf(v,AAc,-2.f*d1) + w*fmaf(w,CCc,-2.f*d2) + 2.f*v*w*ABc;
            dd = fmaxf(dd, 0.f);                    // keep uint ordering valid
            minD[r] = fminf(minD[r], dd);
        }
    }

    // min across the 16 lanes sharing each row (xor shuffles stay in-half)
#pragma unroll
    for (int r=0;r<8;++r){
        float v = minD[r];
        v = fminf(v, __shfl_xor(v, 1, 32));
        v = fminf(v, __shfl_xor(v, 2, 32));
        v = fminf(v, __shfl_xor(v, 4, 32));
        v = fminf(v, __shfl_xor(v, 8, 32));
        minD[r] = v;
    }
    if (m == 0){
        unsigned* mb = minBuf + (size_t)b*P_ + (size_t)ptile*16 + hi*8;
#pragma unroll
        for (int r=0;r<8;++r)
            atomicMin(&mb[r], __float_as_uint(minD[r]));
    }
}

// ---------------- per-batch reduction --------------------------------------
__global__ void reduce_k(const unsigned* __restrict__ minBuf,
                         const float* __restrict__ heights,
                         float* __restrict__ losses){
    __shared__ float sm[256];
    int b = blockIdx.x, tid = threadIdx.x;
    const unsigned* mb = minBuf + (size_t)b*P_;
    float s = 0.f;
    for (int i = tid; i < P_; i += 256) s += __uint_as_float(mb[i]);
    sm[tid] = s; __syncthreads();
    for (int st=128; st>0; st>>=1){
        if (tid < st) sm[tid] += sm[tid+st];
        __syncthreads();
    }
    if (tid==0) losses[b] = sqrtf(sm[0]) / heights[b];
}

__global__ void fin_k(const float* __restrict__ losses, float* __restrict__ out){
    if (threadIdx.x==0){
        float s = 0.f;
        for (int b=0;b<B_;++b) s += losses[b];
        out[0] = s / (float)B_;
    }
}

// ---------------------------------------------------------------------------
extern "C" void kernel_launch(void* const* d_in, const int* in_sizes, int n_in,
                              void* d_out, int out_size, void* d_ws, size_t ws_size,
                              hipStream_t stream){
    (void)in_sizes; (void)n_in; (void)out_size; (void)ws_size;
    const float* points = (const float*)d_in[0];   // [B,P,3]
    const float* verts  = (const float*)d_in[1];   // [B,V,3]
    const int*   faces  = (const int*)  d_in[2];   // [B,F,3]
    float* out = (float*)d_out;

    float*    ws      = (float*)d_ws;
    float*    tri     = ws;                                           // B*F*20 floats
    unsigned* minBuf  = (unsigned*)(ws + (size_t)B_*F_*TRI_STRIDE);   // B*P uints
    float*    heights = (float*)(minBuf + (size_t)B_*P_);             // B floats
    float*    losses  = heights + B_;                                 // B floats

    init_k  <<<(B_*P_+255)/256, 256, 0, stream>>>(minBuf, B_*P_);
    prep_tri<<<(B_*F_+255)/256, 256, 0, stream>>>(verts, faces, tri);
    height_k<<<B_, 128, 0, stream>>>(verts, heights);
    dist_k  <<<dim3(P_/16, SPLIT, B_), 32, 0, stream>>>(points, tri, minBuf);
    reduce_k<<<B_, 256, 0, stream>>>(minBuf, heights, losses);
    fin_k   <<<1, 1, 0, stream>>>(losses, out);
}